// Model_79937931313415
// MI455X (gfx1250) — compile-verified
//
#include <hip/hip_runtime.h>
#include <hip/hip_bf16.h>

typedef __attribute__((ext_vector_type(2))) float v2f;
typedef __attribute__((ext_vector_type(8))) float v8f;

#define NEG_SLOPE 0.2f

// ---------- helpers: monotone float<->ordered-uint map for atomic max ----------
__device__ __forceinline__ unsigned ford(float f) {
    unsigned u = __float_as_uint(f);
    return (u & 0x80000000u) ? ~u : (u | 0x80000000u);
}
__device__ __forceinline__ float ford_inv(unsigned u) {
    return __uint_as_float((u & 0x80000000u) ? (u & 0x7fffffffu) : ~u);
}

// ---------- input encoders: out[i,c] = x[i]*Wl[c] + bl[c] + emb[nid[i],c] ----------
__global__ void enc_kernel(const float* __restrict__ x, const float* __restrict__ Wl,
                           const float* __restrict__ bl, const float* __restrict__ emb,
                           const int* __restrict__ nid, float* __restrict__ out, int n) {
    int t = blockIdx.x * blockDim.x + threadIdx.x;
    if (t >= n * 64) return;
    int i = t >> 6, c = t & 63;
    out[t] = x[i] * Wl[c] + bl[c] + emb[(size_t)nid[i] * 64 + c];
}

// ---------- vad[i] = sum_j Wd[i,j] * ad[j]  (64 threads) ----------
__global__ void matvec64_kernel(const float* __restrict__ W, const float* __restrict__ v,
                                float* __restrict__ out) {
    int i = threadIdx.x;
    float sum = 0.f;
#pragma unroll 8
    for (int j = 0; j < 64; ++j) sum += W[i * 64 + j] * v[j];
    out[i] = sum;
}

// ---------- out[i] = dot(A[i,:], v)  (per-node 64-dot) ----------
__global__ void rowdot_kernel(const float* __restrict__ A, const float* __restrict__ v,
                              float* __restrict__ out, int n) {
    int i = blockIdx.x * blockDim.x + threadIdx.x;
    if (i >= n) return;
    const float4* a4 = reinterpret_cast<const float4*>(A + (size_t)i * 64);
    const float4* v4 = reinterpret_cast<const float4*>(v);
    float sum = 0.f;
#pragma unroll
    for (int k = 0; k < 16; ++k) {
        float4 a = a4[k], b = v4[k];
        sum += a.x * b.x + a.y * b.y + a.z * b.z + a.w * b.w;
    }
    out[i] = sum;
}

// ---------- C[N,64] = A[N,64] @ W[64,64], fp32 WMMA, one 16x16 tile per wave ----------
__global__ void gemm64_wmma(const float* __restrict__ A, const float* __restrict__ W,
                            float* __restrict__ C, int nrows) {
    int wave = (blockIdx.x * blockDim.x + threadIdx.x) >> 5;
    int lane = threadIdx.x & 31;
    int m0 = (wave >> 2) * 16;          // row tile (wave-uniform)
    if (m0 >= nrows) return;            // uniform exit: EXEC stays all-ones for WMMA
    int n0 = (wave & 3) * 16;           // col tile
    int lo = lane & 15, hi = lane >> 4; // hi in {0,1}
    const float* arow = A + (size_t)(m0 + lo) * 64 + 2 * hi;
    v8f acc = {};
#pragma unroll
    for (int k0 = 0; k0 < 64; k0 += 4) {
        v2f af; af[0] = arow[k0];  af[1] = arow[k0 + 1];
        int krow = k0 + 2 * hi;
        v2f bf; bf[0] = W[krow * 64 + n0 + lo]; bf[1] = W[(krow + 1) * 64 + n0 + lo];
        acc = __builtin_amdgcn_wmma_f32_16x16x4_f32(false, af, false, bf,
                                                    (short)0, acc, false, false);
    }
    float* crow = C + n0 + lo;
#pragma unroll
    for (int r = 0; r < 8; ++r)
        crow[(size_t)(m0 + r + 8 * hi) * 64] = acc[r];
}

// ---------- edge pass A: atomic max of leaky_relu(s[src]+t[dst]) into m[dst] ----------
__global__ void edge_max_kernel(const int* __restrict__ src, const int* __restrict__ dst,
                                const float* __restrict__ s, const float* __restrict__ t,
                                unsigned* __restrict__ m, int ne) {
    int e = blockIdx.x * blockDim.x + threadIdx.x;
    if (e >= ne) return;
    int de = dst[e];
    float a = s[src[e]] + t[de];
    a = (a > 0.f) ? a : NEG_SLOPE * a;
    atomicMax(&m[de], ford(a));
}

// ---------- edge pass B: w=exp(alpha-m); den[dst]+=w; out[dst,:]+=w*hs[src,:] ----------
// one wave per edge; lanes cover the 64 channels as float2
__global__ void edge_acc_kernel(const int* __restrict__ src, const int* __restrict__ dst,
                                const float* __restrict__ s, const float* __restrict__ t,
                                const unsigned* __restrict__ m, float* __restrict__ den,
                                const float* __restrict__ hs, float* __restrict__ out, int ne) {
    int lane = threadIdx.x & 31;
    int wid = (blockIdx.x * blockDim.x + threadIdx.x) >> 5;
    int nw = (gridDim.x * blockDim.x) >> 5;
    for (int e = wid; e < ne; e += nw) {
        int se = src[e], de = dst[e];
        float a = s[se] + t[de];
        a = (a > 0.f) ? a : NEG_SLOPE * a;
        float w = __expf(a - ford_inv(m[de]));
        if (lane == 0)
            __hip_atomic_fetch_add(&den[de], w, __ATOMIC_RELAXED, __HIP_MEMORY_SCOPE_AGENT);
        float2 h = *reinterpret_cast<const float2*>(hs + (size_t)se * 64 + lane * 2);
        float* op = out + (size_t)de * 64 + lane * 2;
        __hip_atomic_fetch_add(op,     w * h.x, __ATOMIC_RELAXED, __HIP_MEMORY_SCOPE_AGENT);
        __hip_atomic_fetch_add(op + 1, w * h.y, __ATOMIC_RELAXED, __HIP_MEMORY_SCOPE_AGENT);
    }
}

// ---------- normalize + bias (+optional relu) in place ----------
__global__ void finalize_kernel(float* __restrict__ out, const float* __restrict__ den,
                                const float* __restrict__ b, int n, int do_relu) {
    int t = blockIdx.x * blockDim.x + threadIdx.x;
    if (t >= n * 64) return;
    int i = t >> 6, c = t & 63;
    float d = den[i];
    float v = (d > 0.f) ? out[t] / d : 0.f;
    v += b[c];
    if (do_relu) v = fmaxf(v, 0.f);
    out[t] = v;
}

// ---------- decoder: gather [zs2|zt2] (128) @ W1(128x64)+b1, relu, @W2(64)+b2 ----------
// one wave per 16 edges; 4 accumulators (16x64) via fp32 WMMA; fused projection
__global__ void decoder_wmma(const float* __restrict__ za_base, const float* __restrict__ zb_base,
                             const int* __restrict__ es, const int* __restrict__ ed,
                             const float* __restrict__ W1, const float* __restrict__ bias1,
                             const float* __restrict__ W2, const float* __restrict__ bias2,
                             float* __restrict__ out, int ne) {
    int wave = (blockIdx.x * blockDim.x + threadIdx.x) >> 5;
    int lane = threadIdx.x & 31;
    int m0 = wave * 16;
    if (m0 >= ne) return;              // wave-uniform
    int lo = lane & 15, hi = lane >> 4;
    const float* pa = za_base + (size_t)es[m0 + lo] * 64;
    const float* pb = zb_base + (size_t)ed[m0 + lo] * 64;

    v8f acc[4];
#pragma unroll
    for (int tt = 0; tt < 4; ++tt) acc[tt] = (v8f){};

    for (int k0 = 0; k0 < 128; k0 += 4) {
        const float* zp = (k0 < 64) ? (pa + k0) : (pb + (k0 - 64));
        v2f af; af[0] = zp[2 * hi]; af[1] = zp[2 * hi + 1];
        int krow = k0 + 2 * hi;
#pragma unroll
        for (int tt = 0; tt < 4; ++tt) {
            v2f bf;
            bf[0] = W1[krow * 64 + tt * 16 + lo];
            bf[1] = W1[(krow + 1) * 64 + tt * 16 + lo];
            acc[tt] = __builtin_amdgcn_wmma_f32_16x16x4_f32(false, af, false, bf,
                                                            (short)0, acc[tt], false, false);
        }
    }

    float b1v[4], w2v[4];
#pragma unroll
    for (int tt = 0; tt < 4; ++tt) { b1v[tt] = bias1[tt * 16 + lo]; w2v[tt] = W2[tt * 16 + lo]; }
    float b2v = bias2[0];

#pragma unroll
    for (int r = 0; r < 8; ++r) {
        float vsum = 0.f;
#pragma unroll
        for (int tt = 0; tt < 4; ++tt) {
            float v = acc[tt][r] + b1v[tt];
            v = fmaxf(v, 0.f);
            vsum += v * w2v[tt];
        }
#pragma unroll
        for (int off = 1; off < 16; off <<= 1)
            vsum += __shfl_xor(vsum, off, 32);
        if (lo == 0) out[m0 + r + 8 * hi] = vsum + b2v;
    }
}

extern "C" void kernel_launch(void* const* d_in, const int* in_sizes, int n_in,
                              void* d_out, int out_size, void* d_ws, size_t ws_size,
                              hipStream_t stream) {
    const float* x_sotu  = (const float*)d_in[0];
    const float* x_taxon = (const float*)d_in[1];
    const int*   nid_s   = (const int*)d_in[2];
    const int*   nid_t   = (const int*)d_in[3];
    const int*   e_src   = (const int*)d_in[4];
    const int*   e_dst   = (const int*)d_in[5];
    const int*   el_src  = (const int*)d_in[6];
    const int*   el_dst  = (const int*)d_in[7];
    const float* Wsl = (const float*)d_in[8];
    const float* bsl = (const float*)d_in[9];
    const float* Wtl = (const float*)d_in[10];
    const float* btl = (const float*)d_in[11];
    const float* emb_s = (const float*)d_in[12];
    const float* emb_t = (const float*)d_in[13];
    // conv params: Ws, Wd, as, ad, b
    const float* P[4][5];
    for (int c = 0; c < 4; ++c)
        for (int j = 0; j < 5; ++j)
            P[c][j] = (const float*)d_in[14 + c * 5 + j];
    const float* dW1 = (const float*)d_in[34];
    const float* db1 = (const float*)d_in[35];
    const float* dW2 = (const float*)d_in[36];
    const float* db2 = (const float*)d_in[37];

    const int NS = in_sizes[0], NT = in_sizes[1], E = in_sizes[4], EL = in_sizes[6];

    // workspace layout (floats)
    float* ws = (float*)d_ws;
    size_t o = 0;
    float* xs  = ws + o; o += (size_t)NS * 64;   // later reused as zs2
    float* xt  = ws + o; o += (size_t)NT * 64;   // later reused as zt2
    float* zs  = ws + o; o += (size_t)NS * 64;
    float* zt  = ws + o; o += (size_t)NT * 64;
    float* hsb = ws + o; o += (size_t)NS * 64;   // NS >= NT
    float* sbuf = ws + o; o += NS;
    float* tbuf = ws + o; o += NS;
    unsigned* mbuf = (unsigned*)(ws + o); o += NS;
    float* denb = ws + o; o += NS;
    float* vad  = ws + o; o += 64;

    // input encoders
    enc_kernel<<<(NS * 64 + 255) / 256, 256, 0, stream>>>(x_sotu, Wsl, bsl, emb_s, nid_s, xs, NS);
    enc_kernel<<<(NT * 64 + 255) / 256, 256, 0, stream>>>(x_taxon, Wtl, btl, emb_t, nid_t, xt, NT);

    auto conv = [&](const float* hsrc, int nsrc, const int* sidx,
                    const float* hdst, int ndst, const int* didx,
                    const float* const* prm, float* outb, int relu) {
        const float* Wsm = prm[0]; const float* Wdm = prm[1];
        const float* as_ = prm[2]; const float* ad_ = prm[3]; const float* bb = prm[4];
        int waves = ((nsrc + 15) / 16) * 4;
        gemm64_wmma<<<(waves * 32 + 255) / 256, 256, 0, stream>>>(hsrc, Wsm, hsb, nsrc);
        rowdot_kernel<<<(nsrc + 255) / 256, 256, 0, stream>>>(hsb, as_, sbuf, nsrc);
        matvec64_kernel<<<1, 64, 0, stream>>>(Wdm, ad_, vad);
        rowdot_kernel<<<(ndst + 255) / 256, 256, 0, stream>>>(hdst, vad, tbuf, ndst);
        hipMemsetAsync(mbuf, 0, (size_t)ndst * 4, stream);          // ordered-key 0 == -NaN < all
        hipMemsetAsync(denb, 0, (size_t)ndst * 4, stream);
        hipMemsetAsync(outb, 0, (size_t)ndst * 64 * 4, stream);
        edge_max_kernel<<<(E + 255) / 256, 256, 0, stream>>>(sidx, didx, sbuf, tbuf, mbuf, E);
        edge_acc_kernel<<<2048, 256, 0, stream>>>(sidx, didx, sbuf, tbuf, mbuf, denb, hsb, outb, E);
        finalize_kernel<<<(ndst * 64 + 255) / 256, 256, 0, stream>>>(outb, denb, bb, ndst, relu);
    };

    // layer 1 (+relu)
    conv(xs, NS, e_src, xt, NT, e_dst, P[0], zt, 1);   // c1st: sotu->taxon
    conv(xt, NT, e_dst, xs, NS, e_src, P[1], zs, 1);   // c1ts: taxon->sotu
    // layer 2 (no relu); outputs alias the now-dead xs/xt buffers
    conv(zs, NS, e_src, zt, NT, e_dst, P[2], xt, 0);   // zt2 := xt buffer
    conv(zt, NT, e_dst, zs, NS, e_src, P[3], xs, 0);   // zs2 := xs buffer

    // decoder: concat-gather + MLP fused into one WMMA kernel
    int dwaves = (EL + 15) / 16;
    decoder_wmma<<<(dwaves * 32 + 255) / 256, 256, 0, stream>>>(
        xs, xt, el_src, el_dst, dW1, db1, dW2, db2, (float*)d_out, EL);
}